// _RiemannianConvDrift_32263794328073
// MI455X (gfx1250) — compile-verified
//
#include <hip/hip_runtime.h>
#include <hip/hip_bf16.h>
#include <stdint.h>

// RiemannianConvDrift for MI455X (gfx1250, wave32).
//
// Pipeline:
//   K1: fused row-normalize + GEMM (yp @ W) via V_WMMA_F32_16X16X4_F32,
//       y tile staged into LDS by the Tensor Data Mover (tensor_load_to_lds)
//   K2: node->edge segment sum (gather xw[inc_nodes], L2 atomics into e_sum/e_deg)
//   K3: e_feat = e_sum / max(e_deg,1)   (in place)
//   K4: edge->node segment sum (gather e_feat[inc_edges], atomics into d_out/n_deg)
//   K5: out = relu(out / max(n_deg,1) + b)
//
// Roofline: ~1.9 GB of gather/stream traffic -> ~80us floor at 23.3 TB/s;
// in practice xw (51.2 MB) and e_sum (25.6 MB) are L2-resident (192 MB L2),
// so the random gathers and the 400M f32 atomics largely resolve in L2.
// The GEMM is 3.3 GFLOP (negligible); fp32 WMMA keeps reference precision.

#define DIMS 128
#define ROWS_PER_BLOCK 16
#define EPSN 1e-6f

typedef __attribute__((ext_vector_type(2))) float    v2f;
typedef __attribute__((ext_vector_type(8))) float    v8f;
typedef __attribute__((ext_vector_type(4))) unsigned v4u;
typedef __attribute__((ext_vector_type(8))) int      v8i_t;
typedef __attribute__((ext_vector_type(4))) int      v4i_t;

#if __has_builtin(__builtin_amdgcn_tensor_load_to_lds)
#define HAVE_TDM 1
#else
#define HAVE_TDM 0
#endif

// ---------------------------------------------------------------------------
// K1: TDM-stage 16 rows of y into LDS, normalize in place, then 8 waves each
//     compute a 16x16 tile of xw = yp @ W with fp32 WMMA (32 x 16x16x4 steps).
// ---------------------------------------------------------------------------
__global__ __launch_bounds__(256) void norm_gemm_kernel(
    const float* __restrict__ y, const float* __restrict__ W,
    float* __restrict__ xw, int n_nodes) {
  __shared__ float lds[ROWS_PER_BLOCK * DIMS];   // 8 KB of 320 KB/WGP
  __shared__ float inv_norm[ROWS_PER_BLOCK];

  const int tid  = threadIdx.x;
  const int lane = tid & 31;
  const int wave = tid >> 5;
  const int row0 = blockIdx.x * ROWS_PER_BLOCK;
  const int rows_avail = min(ROWS_PER_BLOCK, n_nodes - row0);

#if HAVE_TDM
  // --- Tensor Data Mover: one 2D tile (128 elems x 16 rows, 4B elements) ---
  // D# per CDNA5 ISA. tensor_dim1 = rows_avail so a
  // ragged tail reads back zeros (OOB loads return 0).
  if (wave == 0) {
    const unsigned lds_base = (unsigned)(uintptr_t)(&lds[0]); // flat[31:0] == LDS byte offset
    const unsigned long long ga =
        (unsigned long long)(uintptr_t)(y + (size_t)row0 * DIMS);

    v4u g0;
    g0[0] = 1u;                                   // count=1 (valid user D#)
    g0[1] = lds_base;                             // lds_addr
    g0[2] = (unsigned)(ga & 0xFFFFFFFFu);         // global_addr[31:0]
    g0[3] = (unsigned)((ga >> 32) & 0x01FFFFFFu)  // global_addr[56:32]
            | (2u << 30);                         // type = 2 ("image")

    v8i_t g1 = {};
    g1[0] = 0x00020000;                 // data_size=2 (4B), workgroup_mask=0
    g1[1] = (DIMS & 0xFFFF) << 16;      // tensor_dim0[15:0] = 128
    g1[2] = (rows_avail & 0xFFFF) << 16;// tensor_dim0[31:16]=0 | tensor_dim1[15:0]
    g1[3] = (DIMS & 0xFFFF) << 16;      // tensor_dim1[31:16]=0 | tile_dim0 = 128
    g1[4] = ROWS_PER_BLOCK;             // tile_dim1 = 16, tile_dim2 = 0
    g1[5] = DIMS;                       // tensor_dim0_stride[31:0] = 128
    g1[6] = 0;                          // stride hi / tensor_dim1_stride lo
    g1[7] = 0;

    v4i_t gz4 = {};
    v8i_t gz8 = {};
    // clang-23 / therock signature: (v4u, v8i, v4i, v4i, v8i, i32 cpol)
    __builtin_amdgcn_tensor_load_to_lds(g0, g1, gz4, gz4, gz8, 0);
    __builtin_amdgcn_s_wait_tensorcnt(0);
  }
  __syncthreads();
#else
  // Fallback: cooperative float4 load (clamped rows for ragged tail).
  for (int v = tid; v < ROWS_PER_BLOCK * DIMS / 4; v += 256) {
    int r = v >> 5;
    int src_r = min(row0 + r, n_nodes - 1);
    ((float4*)lds)[v] = ((const float4*)(y + (size_t)src_r * DIMS))[v & 31];
  }
  __syncthreads();
#endif

  // Row norms: wave w reduces rows 2w and 2w+1 (float4 per lane + shfl tree).
  #pragma unroll
  for (int rr = 0; rr < 2; ++rr) {
    int r = wave * 2 + rr;
    float4 v = ((const float4*)(lds + r * DIMS))[lane];
    float s = v.x * v.x + v.y * v.y + v.z * v.z + v.w * v.w;
    #pragma unroll
    for (int off = 16; off > 0; off >>= 1) s += __shfl_down(s, off, 32);
    if (lane == 0) inv_norm[r] = 1.0f / fmaxf(sqrtf(s), EPSN);
  }
  __syncthreads();
  for (int e = tid; e < ROWS_PER_BLOCK * DIMS; e += 256)
    lds[e] *= inv_norm[e >> 7];
  __syncthreads();

  // Each wave owns N-tile [n0, n0+16). ISA 7.12.2 fp32 fragment layouts:
  //   A 16x4:  lanes 0-15 -> M=lane, K={kk,kk+1}; lanes 16-31 -> K={kk+2,kk+3}
  //   B 4x16:  lanes 0-15 -> N=n0+lane, K={kk,kk+1}; lanes 16-31 -> K={kk+2,kk+3}
  //   C 16x16: VGPR r, lanes 0-15 -> M=r; lanes 16-31 -> M=r+8
  const int n0    = wave * 16;
  const int lhalf = lane >> 4;
  const int l15   = lane & 15;

  v8f acc = {};
  #pragma unroll 8
  for (int kk = 0; kk < DIMS; kk += 4) {
    const int ka = kk + lhalf * 2;
    v2f a, bf;
    a.x  = lds[l15 * DIMS + ka];
    a.y  = lds[l15 * DIMS + ka + 1];
    bf.x = W[(size_t)(ka + 0) * DIMS + n0 + l15];
    bf.y = W[(size_t)(ka + 1) * DIMS + n0 + l15];
    acc = __builtin_amdgcn_wmma_f32_16x16x4_f32(
        /*neg_a=*/false, a, /*neg_b=*/false, bf,
        /*c_mod=*/(short)0, acc, /*reuse_a=*/false, /*reuse_b=*/false);
  }

  // Uniform full/tail split keeps the hot path free of per-store EXEC churn.
  float* out = xw + (size_t)row0 * DIMS + n0;
  if (rows_avail == ROWS_PER_BLOCK) {
    #pragma unroll
    for (int r = 0; r < 8; ++r) {
      int m = lhalf ? (r + 8) : r;
      out[(size_t)m * DIMS + l15] = acc[r];
    }
  } else {
    #pragma unroll
    for (int r = 0; r < 8; ++r) {
      int m = lhalf ? (r + 8) : r;
      if (m < rows_avail) out[(size_t)m * DIMS + l15] = acc[r];
    }
  }
}

// ---------------------------------------------------------------------------
// K2/K4: one wave per nonzero: gather a 128-f32 row (float4/lane), atomic-add
//        into the segment accumulator (L2-resident -> L2 atomics).
// ---------------------------------------------------------------------------
__global__ __launch_bounds__(256) void segment_scatter_kernel(
    const float* __restrict__ src, const int* __restrict__ gather_idx,
    const int* __restrict__ scatter_idx, float* __restrict__ dst_sum,
    float* __restrict__ dst_deg, int nnz) {
  int i = blockIdx.x * 8 + (threadIdx.x >> 5);
  if (i >= nnz) return;
  const int lane = threadIdx.x & 31;
  const int g = gather_idx[i];
  const int s = scatter_idx[i];
  float4 v = ((const float4*)(src + (size_t)g * DIMS))[lane];
  float* dp = dst_sum + (size_t)s * DIMS + lane * 4;
  atomicAdd(dp + 0, v.x);
  atomicAdd(dp + 1, v.y);
  atomicAdd(dp + 2, v.z);
  atomicAdd(dp + 3, v.w);
  if (lane == 0) atomicAdd(dst_deg + s, 1.0f);
}

// K3: sum /= max(deg, 1) in place (float4 granularity).
__global__ __launch_bounds__(256) void div_deg_kernel(
    float* __restrict__ sum, const float* __restrict__ deg, int rows) {
  int idx = blockIdx.x * 256 + threadIdx.x;      // float4 index
  if (idx >= rows * (DIMS / 4)) return;
  int row = idx >> 5;
  float inv = 1.0f / fmaxf(deg[row], 1.0f);
  float4 v = ((float4*)sum)[idx];
  v.x *= inv; v.y *= inv; v.z *= inv; v.w *= inv;
  ((float4*)sum)[idx] = v;
}

// K5: out = relu(out / max(deg,1) + b)
__global__ __launch_bounds__(256) void finalize_kernel(
    float* __restrict__ out, const float* __restrict__ deg,
    const float* __restrict__ b, int rows) {
  int idx = blockIdx.x * 256 + threadIdx.x;      // float4 index
  if (idx >= rows * (DIMS / 4)) return;
  int row = idx >> 5;
  int c4  = idx & 31;
  float inv = 1.0f / fmaxf(deg[row], 1.0f);
  float4 v  = ((float4*)out)[idx];
  float4 bb = ((const float4*)b)[c4];
  v.x = fmaxf(v.x * inv + bb.x, 0.0f);
  v.y = fmaxf(v.y * inv + bb.y, 0.0f);
  v.z = fmaxf(v.z * inv + bb.z, 0.0f);
  v.w = fmaxf(v.w * inv + bb.w, 0.0f);
  ((float4*)out)[idx] = v;
}

extern "C" void kernel_launch(void* const* d_in, const int* in_sizes, int n_in,
                              void* d_out, int out_size, void* d_ws, size_t ws_size,
                              hipStream_t stream) {
  // inputs: t, y[n,128], W[128,128], b[128], inc_nodes[nnz], inc_edges[nnz]
  const float* y         = (const float*)d_in[1];
  const float* W         = (const float*)d_in[2];
  const float* b         = (const float*)d_in[3];
  const int*   inc_nodes = (const int*)d_in[4];
  const int*   inc_edges = (const int*)d_in[5];

  const int n_nodes = in_sizes[1] / DIMS;   // 100000
  const int nnz     = in_sizes[4];          // 1600000
  const int n_edges = 50000;                // reference constant (not derivable)

  // Workspace partition (aligned 256B): xw 51.2MB + e_sum 25.6MB + degrees.
  char*  ws  = (char*)d_ws;
  size_t off = 0;
  auto carve = [&](size_t bytes) -> void* {
    void* p = ws + off;
    off = (off + bytes + 255) & ~(size_t)255;
    return p;
  };
  float* xw    = (float*)carve((size_t)n_nodes * DIMS * sizeof(float));
  float* e_sum = (float*)carve((size_t)n_edges * DIMS * sizeof(float));
  float* e_deg = (float*)carve((size_t)n_edges * sizeof(float));
  float* n_deg = (float*)carve((size_t)n_nodes * sizeof(float));

  // Zero accumulators (graph-capturable memset nodes).
  (void)hipMemsetAsync(e_sum, 0, (size_t)n_edges * DIMS * sizeof(float), stream);
  (void)hipMemsetAsync(e_deg, 0, (size_t)n_edges * sizeof(float), stream);
  (void)hipMemsetAsync(n_deg, 0, (size_t)n_nodes * sizeof(float), stream);
  (void)hipMemsetAsync(d_out, 0, (size_t)out_size * sizeof(float), stream);

  // K1: normalize + GEMM (100000 rows -> 6250 blocks of 16 rows).
  int gemm_blocks = (n_nodes + ROWS_PER_BLOCK - 1) / ROWS_PER_BLOCK;
  norm_gemm_kernel<<<gemm_blocks, 256, 0, stream>>>(y, W, xw, n_nodes);

  // K2: node -> hyperedge aggregation.
  int sc_blocks = (nnz + 7) / 8;
  segment_scatter_kernel<<<sc_blocks, 256, 0, stream>>>(
      xw, inc_nodes, inc_edges, e_sum, e_deg, nnz);

  // K3: e_feat = e_sum / max(e_deg,1) in place.
  int e_vec = n_edges * (DIMS / 4);
  div_deg_kernel<<<(e_vec + 255) / 256, 256, 0, stream>>>(e_sum, e_deg, n_edges);

  // K4: hyperedge -> node aggregation straight into d_out.
  segment_scatter_kernel<<<sc_blocks, 256, 0, stream>>>(
      e_sum, inc_edges, inc_nodes, (float*)d_out, n_deg, nnz);

  // K5: out = relu(out / max(n_deg,1) + b).
  int n_vec = n_nodes * (DIMS / 4);
  finalize_kernel<<<(n_vec + 255) / 256, 256, 0, stream>>>(
      (float*)d_out, n_deg, b, n_nodes);
}